// FeaSelect_9182640079369
// MI455X (gfx1250) — compile-verified
//
#include <hip/hip_runtime.h>
#include <math.h>

// Problem constants (match reference: B=64, L=4096, D=1024, fp32).
constexpr int Bc = 64;
constexpr int Lc = 4096;
constexpr int Dc = 1024;
constexpr float FILL = -10000.0f;

// Split L into chunks so the grid covers the 23.3 TB/s bandwidth-latency
// product (2048 blocks x 8 wave32s each, 8 B128 loads in flight per thread).
constexpr int ROWCHUNKS = 32;
constexpr int ROWS_PER_CHUNK = Lc / ROWCHUNKS; // 128

typedef float floatx4 __attribute__((ext_vector_type(4)));

__device__ __forceinline__ floatx4 vmax4(floatx4 a, floatx4 b) {
  floatx4 r;
  r.x = __builtin_fmaxf(a.x, b.x);
  r.y = __builtin_fmaxf(a.y, b.y);
  r.z = __builtin_fmaxf(a.z, b.z);
  r.w = __builtin_fmaxf(a.w, b.w);
  return r;
}

// Native CDNA5 device-scope float atomic max (no-return form, STOREcnt).
// scope:SCOPE_DEV is mandatory: partials for one output element come from up
// to 32 different workgroups, so the RMW must be coherent at L2/device scope.
__device__ __forceinline__ void atomic_max_f32_dev(float* p, float v) {
  asm volatile("global_atomic_max_num_f32 %0, %1, off scope:SCOPE_DEV"
               :
               : "v"(p), "v"(v)
               : "memory");
}

// Streaming masked max. Block (b, chunk) covers rows
// [chunk*128, min(chunk*128+128, len)) across the full D=1024 row
// (256 threads x float4). Non-temporal B128 loads (one-pass data, bypass L2
// retention), 8 loads in flight per thread, then one native float atomic max
// per output element.
__global__ __launch_bounds__(256) void fea_maxpool_reduce(
    const float* __restrict__ feat, const int* __restrict__ lengths,
    float* __restrict__ out) {
  const int b     = blockIdx.x;
  const int chunk = blockIdx.y;
  const int len   = lengths[b];
  const int l0    = chunk * ROWS_PER_CHUNK;
  int nrows = len - l0;
  if (nrows <= 0) return;
  if (nrows > ROWS_PER_CHUNK) nrows = ROWS_PER_CHUNK;

  const int col = threadIdx.x << 2;                 // 0..1020, step 4
  const floatx4* p = (const floatx4*)(feat + (size_t)b * Lc * Dc
                                           + (size_t)l0 * Dc + col);
  constexpr int RS = Dc / 4;                        // row stride in floatx4

  floatx4 acc0 = (floatx4)(-INFINITY);
  floatx4 acc1 = (floatx4)(-INFINITY);

  int r = 0;
  for (; r + 8 <= nrows; r += 8) {
    floatx4 v0 = __builtin_nontemporal_load(p + 0 * RS);
    floatx4 v1 = __builtin_nontemporal_load(p + 1 * RS);
    floatx4 v2 = __builtin_nontemporal_load(p + 2 * RS);
    floatx4 v3 = __builtin_nontemporal_load(p + 3 * RS);
    floatx4 v4 = __builtin_nontemporal_load(p + 4 * RS);
    floatx4 v5 = __builtin_nontemporal_load(p + 5 * RS);
    floatx4 v6 = __builtin_nontemporal_load(p + 6 * RS);
    floatx4 v7 = __builtin_nontemporal_load(p + 7 * RS);
    acc0 = vmax4(acc0, vmax4(vmax4(v0, v1), vmax4(v2, v3)));
    acc1 = vmax4(acc1, vmax4(vmax4(v4, v5), vmax4(v6, v7)));
    p += 8 * RS;
  }
  for (; r < nrows; ++r) {
    acc0 = vmax4(acc0, __builtin_nontemporal_load(p));
    p += RS;
  }
  floatx4 acc = vmax4(acc0, acc1);

  float* o = out + (size_t)b * Dc + col;
  atomic_max_f32_dev(o + 0, acc.x);
  atomic_max_f32_dev(o + 1, acc.y);
  atomic_max_f32_dev(o + 2, acc.z);
  atomic_max_f32_dev(o + 3, acc.w);
}

// Initialize output with the exact reference identity element.
//   len == 0 : result forced to 0 (reference zeroes empty sequences)
//   len <  L : at least one masked slot holds FILL, so FILL floors the max
//   len == L : no masked slot; identity is -inf
__global__ __launch_bounds__(256) void fea_maxpool_init(
    const int* __restrict__ lengths, float* __restrict__ out) {
  const int idx = blockIdx.x * 256 + threadIdx.x;   // over B*D = 65536
  const int b   = idx >> 10;                        // / Dc
  const int len = lengths[b];
  float v = (len == 0) ? 0.0f : ((len < Lc) ? FILL : -INFINITY);
  out[idx] = v;
}

extern "C" void kernel_launch(void* const* d_in, const int* in_sizes, int n_in,
                              void* d_out, int out_size, void* d_ws, size_t ws_size,
                              hipStream_t stream) {
  (void)in_sizes; (void)n_in; (void)out_size; (void)d_ws; (void)ws_size;
  const float* feat    = (const float*)d_in[0];   // [B, L, D] fp32
  const int*   lengths = (const int*)d_in[1];     // [B] int32
  float*       out     = (float*)d_out;           // [B, D] fp32

  fea_maxpool_init<<<(Bc * Dc) / 256, 256, 0, stream>>>(lengths, out);

  dim3 grid(Bc, ROWCHUNKS);
  fea_maxpool_reduce<<<grid, 256, 0, stream>>>(feat, lengths, out);
}